// SumNode_11905649344609
// MI455X (gfx1250) — compile-verified
//
#include <hip/hip_runtime.h>

// Sorted-segment row-sum: feat[N,128] f32 -> out[G,128] f32.
//
// Roofline: 512 MB streamed read + 2 MB write @ 23.3 TB/s => ~22 us floor;
// 128M adds are negligible (0.25 FLOP/byte) -> pure memory-path problem.
//
// Phase 1: bounds[g] = lower_bound(segment_ids, g) for g in [0, G], computed
//          by 4097 parallel threads (one ~20-deep load chain total, not one
//          per block).
// Phase 2: one workgroup per segment streams its contiguous row range with
//          non-temporal b128 loads (+ global_prefetch_b8), accumulates in
//          VGPRs, tiny deterministic LDS tree across the 8 waves, b128 store.
// No atomics anywhere -> bitwise-deterministic output.

typedef float v4f __attribute__((ext_vector_type(4)));

#define DFEAT 128
#define COLS4 32          // 128 floats = 32 float4 slices
#define ROWP  8           // row-parallel groups (256 threads / 32 lanes)
#define BLOCK 256

__global__ __launch_bounds__(BLOCK) void seg_bounds_kernel(
    const int* __restrict__ seg,
    int*       __restrict__ bounds,
    int n_nodes, int num_segments)
{
    const int g = (int)blockIdx.x * BLOCK + (int)threadIdx.x;
    if (g > num_segments) return;
    if (g == num_segments) { bounds[g] = n_nodes; return; }
    int lo = 0, hi = n_nodes;
    while (lo < hi) { int m = (lo + hi) >> 1; if (seg[m] < g) lo = m + 1; else hi = m; }
    bounds[g] = lo;
}

__global__ __launch_bounds__(BLOCK) void segsum_f32_kernel(
    const float* __restrict__ feat,
    const int*   __restrict__ seg,     // used only if bounds == nullptr
    const int*   __restrict__ bounds,  // precomputed segment boundaries (or null)
    float*       __restrict__ out,
    int n_nodes)
{
    const int g    = (int)blockIdx.x;
    const int tid  = (int)threadIdx.x;
    const int col4 = tid & (COLS4 - 1);   // which 16B column slice
    const int rowp = tid >> 5;            // which row-group (0..7)

    int row_start, row_end;
    if (bounds) {
        // two uniform scalar loads; no serial search on the critical path
        row_start = bounds[g];
        row_end   = bounds[g + 1];
    } else {
        int lo = 0, hi = n_nodes;
        while (lo < hi) { int m = (lo + hi) >> 1; if (seg[m] <  g) lo = m + 1; else hi = m; }
        row_start = lo;
        hi = n_nodes;
        while (lo < hi) { int m = (lo + hi) >> 1; if (seg[m] <= g) lo = m + 1; else hi = m; }
        row_end = lo;
    }
    const int nrows = row_end - row_start;

    const float* base = feat + (size_t)row_start * DFEAT + (size_t)col4 * 4;

    v4f a0 = {0.f, 0.f, 0.f, 0.f};
    v4f a1 = {0.f, 0.f, 0.f, 0.f};
    v4f a2 = {0.f, 0.f, 0.f, 0.f};
    v4f a3 = {0.f, 0.f, 0.f, 0.f};

    int r = rowp;
    // 4-way unrolled: 4 independent non-temporal b128 loads in flight per lane
    for (; r + 3 * ROWP < nrows; r += 4 * ROWP) {
        const float* p = base + (size_t)r * DFEAT;
        // prefetch 2 unroll-iterations ahead (guarded in-bounds of this segment)
        if (r + 8 * ROWP < nrows)
            __builtin_prefetch(p + (size_t)8 * ROWP * DFEAT, 0, 0);
        v4f v0 = __builtin_nontemporal_load((const v4f*)(p));
        v4f v1 = __builtin_nontemporal_load((const v4f*)(p + 1 * ROWP * DFEAT));
        v4f v2 = __builtin_nontemporal_load((const v4f*)(p + 2 * ROWP * DFEAT));
        v4f v3 = __builtin_nontemporal_load((const v4f*)(p + 3 * ROWP * DFEAT));
        a0 += v0; a1 += v1; a2 += v2; a3 += v3;
    }
    for (; r < nrows; r += ROWP)
        a0 += __builtin_nontemporal_load((const v4f*)(base + (size_t)r * DFEAT));

    a0 += a1; a2 += a3; a0 += a2;   // fixed order -> deterministic

    // cross-wave reduction over the 8 row-groups (4 KB LDS, b128 accesses)
    __shared__ v4f smem[ROWP][COLS4];
    smem[rowp][col4] = a0;
    __syncthreads();
    #pragma unroll
    for (int s = ROWP >> 1; s > 0; s >>= 1) {
        if (rowp < s) smem[rowp][col4] += smem[rowp + s][col4];
        __syncthreads();
    }
    if (rowp == 0) {
        *(v4f*)(out + (size_t)g * DFEAT + (size_t)col4 * 4) = smem[0][col4];
    }
}

extern "C" void kernel_launch(void* const* d_in, const int* in_sizes, int n_in,
                              void* d_out, int out_size, void* d_ws, size_t ws_size,
                              hipStream_t stream) {
    const float* feat = (const float*)d_in[0];
    const int*   seg  = (const int*)d_in[1];
    const int num_segments = out_size / DFEAT;   // 4096
    const int n_nodes      = in_sizes[1];        // 1,000,000

    const size_t bounds_bytes = (size_t)(num_segments + 1) * sizeof(int);
    int* bounds = (ws_size >= bounds_bytes) ? (int*)d_ws : nullptr;

    if (bounds) {
        const int nb = (num_segments + 1 + BLOCK - 1) / BLOCK;
        seg_bounds_kernel<<<nb, BLOCK, 0, stream>>>(seg, bounds, n_nodes, num_segments);
    }
    segsum_f32_kernel<<<num_segments, BLOCK, 0, stream>>>(feat, seg, bounds,
                                                          (float*)d_out, n_nodes);
}